// MLPModel_11879879543290
// MI455X (gfx1250) — compile-verified
//
#include <hip/hip_runtime.h>
#include <hip/hip_bf16.h>

// ---------------- problem constants ----------------
#define NV        200000            // active voxels (divisible by 16)
#define NT        (NV/16)           // 12500 row-tiles
#define DD        41
#define HH        1600
#define WW        1408
#define HASH_BITS 20
#define HASH_SIZE (1<<HASH_BITS)
#define HASH_MASK (HASH_SIZE-1)
#define BN_EPS    0.001f
#define NBLK_STATS 256
#define NBS       28               // rulebook stride: 27 offsets + 1 pad slot

typedef float v2f __attribute__((ext_vector_type(2)));
typedef float v8f __attribute__((ext_vector_type(8)));

// ---- CDNA5 async global->LDS copy (ASYNCcnt-tracked), per CDNA5 ISA
__device__ __forceinline__ void async_g2l_b32(void* lds_ptr, const void* gptr) {
  unsigned lds_addr = (unsigned)(size_t)lds_ptr;   // LDS aperture: low 32 bits = LDS offset
  asm volatile("global_load_async_to_lds_b32 %0, %1, off"
               :: "v"(lds_addr), "v"(gptr) : "memory");
}
__device__ __forceinline__ void wait_async0() {
  asm volatile("s_wait_asynccnt 0x0" ::: "memory");
}

// ---------------- hash grid ----------------
__device__ __forceinline__ unsigned hash_fn(int flat) {
  return ((unsigned)flat * 2654435761u) & HASH_MASK;
}

__global__ void k_hash_init(int* __restrict__ hkey) {
  int i = blockIdx.x * blockDim.x + threadIdx.x;
  if (i < HASH_SIZE) hkey[i] = -1;
}

__global__ void k_hash_insert(const int* __restrict__ coords,
                              int* __restrict__ hkey, int* __restrict__ hval) {
  int n = blockIdx.x * blockDim.x + threadIdx.x;
  if (n >= NV) return;
  int b = coords[n*4+0], z = coords[n*4+1], y = coords[n*4+2], x = coords[n*4+3];
  int flat = ((b*DD + z)*HH + y)*WW + x;
  unsigned s = hash_fn(flat);
  while (true) {
    int old = atomicCAS(&hkey[s], -1, flat);
    if (old == -1) { hval[s] = n; break; }
    if (old == flat) break;
    s = (s + 1) & HASH_MASK;
  }
}

// rulebook: nbr[n*28+k] = neighbor index or NV sentinel
__global__ void k_build_nbr(const int* __restrict__ coords,
                            const int* __restrict__ hkey,
                            const int* __restrict__ hval,
                            int* __restrict__ nbr) {
  int n = blockIdx.x * blockDim.x + threadIdx.x;
  if (n >= NV) return;
  int b = coords[n*4+0], z = coords[n*4+1], y = coords[n*4+2], x = coords[n*4+3];
  for (int k = 0; k < 27; ++k) {
    int dz = k/9 - 1, dy = (k/3)%3 - 1, dx = k%3 - 1;
    int nz = z+dz, ny = y+dy, nx = x+dx;
    int res = NV;
    if (nz >= 0 && nz < DD && ny >= 0 && ny < HH && nx >= 0 && nx < WW) {
      int flat = ((b*DD + nz)*HH + ny)*WW + nx;
      unsigned s = hash_fn(flat);
      while (true) {
        int key = hkey[s];
        if (key == flat) { res = hval[s]; break; }
        if (key == -1) break;
        s = (s + 1) & HASH_MASK;
      }
    }
    nbr[n*NBS + k] = res;
  }
  nbr[n*NBS + 27] = NV;
}

// features padded to 4 channels (channel 3 = 0), plus zero row at index NV
__global__ void k_padfeat(const float* __restrict__ feat, float* __restrict__ featp4) {
  int i = blockIdx.x * blockDim.x + threadIdx.x;
  if (i >= (NV+1)*4) return;
  int n = i >> 2, c = i & 3;
  featp4[i] = (n < NV && c < 3) ? feat[n*3 + c] : 0.f;
}

// zero row NV of padded activation buffer (16 floats)
__global__ void k_zero_row(float* __restrict__ hp) {
  int i = threadIdx.x;
  if (i < 16) hp[NV*16 + i] = 0.f;
}

// Pre-permute weights into per-lane WMMA B fragments (float2 per lane per K-chunk).
// conv1: 27 chunks over padded K=108 (flat p = k*4 + ch4; ch4==3 is zero pad)
// conv2: 108 chunks over K=432 (flat pp = k*16 + cin)
__global__ void k_prep_frags(const float* __restrict__ w1, const float* __restrict__ w2,
                             v2f* __restrict__ w1f, v2f* __restrict__ w2f) {
  int t = blockIdx.x * blockDim.x + threadIdx.x;
  if (t < 27*32) {
    int k = t >> 5, lane = t & 31;
    int r = lane & 15, half = lane >> 4;
    v2f v;
    v.x = w1[(k*3 + 2*half)*16 + r];                // ch4 = 2*half in {0,2}
    v.y = half ? 0.f : w1[(k*3 + 1)*16 + r];        // ch4 = 2*half+1 in {1,3}; 3 -> pad
    w1f[t] = v;
  }
  int t2 = t - 27*32;
  if (t2 >= 0 && t2 < 108*32) {
    int c = t2 >> 5, lane = t2 & 31;
    int r = lane & 15, half = lane >> 4;
    int pp0 = 4*c + 2*half;
    v2f v;
    v.x = w2[pp0*16 + r];
    v.y = w2[(pp0 + 1)*16 + r];
    w2f[t2] = v;
  }
}

// ---------------- layer 1 conv: (N,4-padded) x w1 via f32 WMMA, 27 K=4 chunks ----------------
__global__ void __launch_bounds__(128) k_conv1(const float* __restrict__ featp4,
                                               const int* __restrict__ nbrx,
                                               const v2f* __restrict__ w1f,
                                               float* __restrict__ y) {
  __shared__ int s_nbr[4 * 16 * NBS];               // 4 tiles x 16 rows x 28 = 1792
  const int tid = threadIdx.x;
  const size_t base = (size_t)blockIdx.x * (4 * 16 * NBS);
  for (int i = tid; i < 4 * 16 * NBS; i += 128)     // 1792 = 14*128, no tail divergence
    async_g2l_b32(&s_nbr[i], &nbrx[base + i]);
  wait_async0();
  __syncthreads();

  const int lane = tid & 31, wave = tid >> 5;
  const int r = lane & 15, half = lane >> 4;
  const int tile = blockIdx.x * 4 + wave;           // NT divisible by 4
  const int* mynbr = &s_nbr[wave * 16 * NBS + r * NBS];

  v2f B[27];
  #pragma unroll
  for (int k = 0; k < 27; ++k) B[k] = w1f[k*32 + lane];

  v8f C = {};
  #pragma unroll
  for (int k = 0; k < 27; ++k) {
    int j = mynbr[k];                               // <= NV; row NV of featp4 is zero
    v2f A = *(const v2f*)(featp4 + j*4 + 2*half);   // aligned float2, branch-free
    C = __builtin_amdgcn_wmma_f32_16x16x4_f32(false, A, false, B[k],
                                              (short)0, C, false, false);
  }

  #pragma unroll
  for (int i = 0; i < 8; ++i) {
    int m = half * 8 + i;                           // D layout: vgpr i -> M = i (+8 hi half)
    y[(tile*16 + m)*16 + r] = C[i];
  }
}

// ---------------- layer 2 conv: (N,16) x w2(27,16,16); 4 waves split 108 K-chunks ----------------
__global__ void __launch_bounds__(128) k_conv2(const float* __restrict__ hp,
                                               const int* __restrict__ nbrx,
                                               const v2f* __restrict__ w2f,
                                               float* __restrict__ y) {
  __shared__ int   s_nbr[16 * NBS];                 // 448 ints
  __shared__ float s_red[3 * 32 * 8];
  const int tid = threadIdx.x;
  const int tile = blockIdx.x;
  const size_t base = (size_t)tile * (16 * NBS);
  for (int i = tid; i < 16 * NBS; i += 128)
    async_g2l_b32(&s_nbr[i], &nbrx[base + i]);
  wait_async0();
  __syncthreads();

  const int lane = tid & 31, wave = tid >> 5;
  const int r = lane & 15, half = lane >> 4;

  v2f B[27];
  #pragma unroll
  for (int cc = 0; cc < 27; ++cc) B[cc] = w2f[(wave*27 + cc)*32 + lane];

  v8f C = {};
  #pragma unroll
  for (int cc = 0; cc < 27; ++cc) {
    int c   = wave*27 + cc;
    int k   = c >> 2;                               // pp0 = 4c+2*half; k = pp0/16 = c/4
    int cin = (c & 3)*4 + 2*half;                   // even -> 8B-aligned float2
    int j   = s_nbr[r*NBS + k];                     // <= NV; row NV of hp is zero
    v2f A = *(const v2f*)(hp + j*16 + cin);
    C = __builtin_amdgcn_wmma_f32_16x16x4_f32(false, A, false, B[cc],
                                              (short)0, C, false, false);
  }

  if (wave != 0) {
    #pragma unroll
    for (int i = 0; i < 8; ++i) s_red[((wave-1)*32 + lane)*8 + i] = C[i];
  }
  __syncthreads();
  if (wave == 0) {
    #pragma unroll
    for (int i = 0; i < 8; ++i) {
      float v = C[i] + s_red[(0*32 + lane)*8 + i]
                     + s_red[(1*32 + lane)*8 + i]
                     + s_red[(2*32 + lane)*8 + i];
      int m = half * 8 + i;
      y[(tile*16 + m)*16 + r] = v;
    }
  }
}

// ---------------- deterministic BN statistics (two-stage) ----------------
__global__ void __launch_bounds__(256) k_stats1(const float* __restrict__ y,
                                                float* __restrict__ part) {
  __shared__ float ssum[256], ssq[256];
  int tid = threadIdx.x;
  int ch = tid & 15, g = tid >> 4;
  int G = blockIdx.x * 16 + g;
  float s = 0.f, q = 0.f;
  for (int r = G; r < NV; r += NBLK_STATS * 16) {
    float v = y[r*16 + ch];
    s += v; q += v * v;
  }
  ssum[tid] = s; ssq[tid] = q;
  __syncthreads();
  if (tid < 16) {
    float S = 0.f, Q = 0.f;
    for (int i = 0; i < 16; ++i) { S += ssum[tid + 16*i]; Q += ssq[tid + 16*i]; }
    part[blockIdx.x*32 + tid]      = S;
    part[blockIdx.x*32 + 16 + tid] = Q;
  }
}

__global__ void k_stats2(const float* __restrict__ part, float* __restrict__ stats) {
  int tid = threadIdx.x;
  if (tid < 16) {
    float S = 0.f, Q = 0.f;
    for (int b = 0; b < NBLK_STATS; ++b) { S += part[b*32 + tid]; Q += part[b*32 + 16 + tid]; }
    float mean = S / (float)NV;
    float var  = Q / (float)NV - mean * mean;       // population variance (jnp.var)
    stats[tid]      = mean;
    stats[16 + tid] = rsqrtf(var + BN_EPS);
  }
}

__global__ void k_bnrelu(const float* __restrict__ y, const float* __restrict__ stats,
                         const float* __restrict__ gamma, const float* __restrict__ beta,
                         float* __restrict__ out) {
  int idx = blockIdx.x * blockDim.x + threadIdx.x;
  if (idx >= NV * 16) return;
  int ch = idx & 15;
  float v = (y[idx] - stats[ch]) * stats[16 + ch] * gamma[ch] + beta[ch];
  out[idx] = v > 0.f ? v : 0.f;
}

// ---------------- host launcher ----------------
extern "C" void kernel_launch(void* const* d_in, const int* in_sizes, int n_in,
                              void* d_out, int out_size, void* d_ws, size_t ws_size,
                              hipStream_t stream) {
  (void)in_sizes; (void)n_in; (void)out_size; (void)ws_size;
  const float* feat   = (const float*)d_in[0];
  const int*   coords = (const int*)  d_in[1];
  const float* w1     = (const float*)d_in[2];
  const float* g1     = (const float*)d_in[3];
  const float* b1     = (const float*)d_in[4];
  const float* w2     = (const float*)d_in[5];
  const float* g2     = (const float*)d_in[6];
  const float* b2     = (const float*)d_in[7];
  float* out = (float*)d_out;

  char* ws = (char*)d_ws;
  size_t off = 0;
  auto alloc = [&](size_t bytes) -> char* {
    char* p = ws + off;
    off = (off + bytes + 255) & ~(size_t)255;
    return p;
  };
  int*   hkey  = (int*)  alloc((size_t)HASH_SIZE * 4);
  int*   hval  = (int*)  alloc((size_t)HASH_SIZE * 4);
  int*   nbrx  = (int*)  alloc((size_t)NV * NBS * 4);
  float* featp4= (float*)alloc((size_t)(NV + 1) * 4 * 4);
  float* ybuf  = (float*)alloc((size_t)NV * 16 * 4);
  float* hp    = (float*)alloc((size_t)(NV + 1) * 16 * 4);
  v2f*   w1f   = (v2f*)  alloc((size_t)27 * 32 * 8);
  v2f*   w2f   = (v2f*)  alloc((size_t)108 * 32 * 8);
  float* part  = (float*)alloc((size_t)NBLK_STATS * 32 * 4);
  float* stats = (float*)alloc(32 * 4);

  hipLaunchKernelGGL(k_hash_init,   dim3(HASH_SIZE/256),       dim3(256), 0, stream, hkey);
  hipLaunchKernelGGL(k_hash_insert, dim3((NV+255)/256),        dim3(256), 0, stream, coords, hkey, hval);
  hipLaunchKernelGGL(k_build_nbr,   dim3((NV+255)/256),        dim3(256), 0, stream, coords, hkey, hval, nbrx);
  hipLaunchKernelGGL(k_padfeat,     dim3(((NV+1)*4+255)/256),  dim3(256), 0, stream, feat, featp4);
  hipLaunchKernelGGL(k_prep_frags,  dim3((135*32+255)/256),    dim3(256), 0, stream, w1, w2, w1f, w2f);

  hipLaunchKernelGGL(k_conv1,  dim3(NT/4),            dim3(128), 0, stream, featp4, nbrx, w1f, ybuf);
  hipLaunchKernelGGL(k_stats1, dim3(NBLK_STATS),      dim3(256), 0, stream, ybuf, part);
  hipLaunchKernelGGL(k_stats2, dim3(1),               dim3(32),  0, stream, part, stats);
  hipLaunchKernelGGL(k_bnrelu, dim3((NV*16+255)/256), dim3(256), 0, stream, ybuf, stats, g1, b1, hp);
  hipLaunchKernelGGL(k_zero_row, dim3(1),             dim3(32),  0, stream, hp);

  hipLaunchKernelGGL(k_conv2,  dim3(NT),              dim3(128), 0, stream, hp, nbrx, w2f, ybuf);
  hipLaunchKernelGGL(k_stats1, dim3(NBLK_STATS),      dim3(256), 0, stream, ybuf, part);
  hipLaunchKernelGGL(k_stats2, dim3(1),               dim3(32),  0, stream, part, stats);
  hipLaunchKernelGGL(k_bnrelu, dim3((NV*16+255)/256), dim3(256), 0, stream, ybuf, stats, g2, b2, out);
}